// DecompGridv6_78099685310708
// MI455X (gfx1250) — compile-verified
//
#include <hip/hip_runtime.h>

typedef float v2f __attribute__((ext_vector_type(2)));
typedef float v4f __attribute__((ext_vector_type(4)));

constexpr int RESO = 512;
constexpr int CH = 32;
constexpr int PLANE_ELEMS = RESO * RESO;          // texels per channel / per plane
constexpr int RSTRIDE = RESO * CH;                // floats per image row in (H,W,C)
constexpr size_t TBYTES = 3ull * CH * PLANE_ELEMS * sizeof(float);  // 96 MB

// ---------- small load helpers ----------
__device__ __forceinline__ v2f ld2(const float* __restrict__ p) {
  v2f r;
  __builtin_memcpy(&r, p, sizeof(r));   // 8B gather (dword-aligned ok on gfx1250)
  return r;
}
__device__ __forceinline__ v4f ld4(const float* __restrict__ p) {
  v4f r;
  __builtin_memcpy(&r, p, sizeof(r));   // 16B-aligned b128 gather
  return r;
}

// One coordinate -> (cell, frac), torch grid_sample align_corners=True clamp
// semantics. Top edge shifted (c=511 -> c=510, w=1) so the contiguous x-pair
// load selects exactly texel 511, matching the reference's double-clamp.
__device__ __forceinline__ void decomp(float g, int& c, float& w) {
  float ix = (g + 1.0f) * (0.5f * (RESO - 1));
  float f = floorf(ix);
  w = ix - f;
  int ci = (int)f;
  ci = ci < 0 ? 0 : ci;
  if (ci > RESO - 2) { ci = RESO - 2; w = 1.0f; }
  c = ci;
}

// ---------- pass 1: (C,H,W) -> (H,W,C) transpose, one 32x-by-32c tile/wave ----
// Reads: NT (originals are dead after this; keep them out of L2).
// Writes: RT (transposed planes become the L2-resident gather working set).
__global__ void __launch_bounds__(256)
transpose_planes_kernel(const float* __restrict__ p0,
                        const float* __restrict__ p1,
                        const float* __restrict__ p2,
                        float* __restrict__ T) {
  __shared__ float tile[8][32 * 36];    // stride 36: 16B-aligned rows, few conflicts
  const int w    = threadIdx.x >> 5;
  const int lane = threadIdx.x & 31;
  const int wid  = blockIdx.x * 8 + w;  // 0 .. 3*512*16 - 1
  const int pl   = wid >> 13;           // / (512*16)
  const int rem  = wid & 8191;
  const int y    = rem >> 4;
  const int xb   = (rem & 15) << 5;

  const float* __restrict__ src = (pl == 0) ? p0 : (pl == 1) ? p1 : p2;
  const float* __restrict__ s = src + (y * RESO + xb + lane);
#pragma unroll
  for (int c = 0; c < CH; ++c)
    tile[w][lane * 36 + c] =
        __builtin_nontemporal_load(s + (size_t)c * PLANE_ELEMS);

  __syncthreads();

  float* __restrict__ d =
      T + (((size_t)pl * PLANE_ELEMS + (size_t)(y * RESO + xb + lane)) << 5);
#pragma unroll
  for (int i = 0; i < 8; ++i) {
    v4f v;
    __builtin_memcpy(&v, &tile[w][lane * 36 + 4 * i], 16);
    *(v4f*)(d + 4 * i) = v;             // regular-temporal: want this in L2
  }
}

// ---------- pass 2: gather + bilinear + product on (H,W,C) layout ----------
__global__ void __launch_bounds__(256)
triplane_sample_tr_kernel(const float* __restrict__ x,
                          const float* __restrict__ T,
                          float* __restrict__ out, int n) {
  int p = blockIdx.x * 256 + threadIdx.x;
  if (p >= n) return;

  const float* __restrict__ xp = x + 3ull * (unsigned)p;
  float g0 = __builtin_nontemporal_load(xp + 0);
  float g1 = __builtin_nontemporal_load(xp + 1);
  float g2 = __builtin_nontemporal_load(xp + 2);

  int c0, c1, c2;
  float w0, w1, w2;
  decomp(g0, c0, w0);
  decomp(g1, c1, w1);
  decomp(g2, c2, w2);

  // One 64-bit base per plane; all 4 corners + channel chunk reachable via
  // 24-bit immediate offsets (row step 64KB, x step 128B, chunk step 16B).
  const float* __restrict__ b0 = T + ((size_t)(c1 * RESO + c0) << 5);
  const float* __restrict__ b1 =
      T + (((size_t)PLANE_ELEMS + (size_t)(c2 * RESO + c0)) << 5);
  const float* __restrict__ b2 =
      T + (((size_t)2 * PLANE_ELEMS + (size_t)(c2 * RESO + c1)) << 5);

  const float a00 = (1.0f - w0) * (1.0f - w1), a01 = w0 * (1.0f - w1);
  const float a10 = (1.0f - w0) * w1,          a11 = w0 * w1;
  const float b00 = (1.0f - w0) * (1.0f - w2), b01 = w0 * (1.0f - w2);
  const float b10 = (1.0f - w0) * w2,          b11 = w0 * w2;
  const float c00 = (1.0f - w1) * (1.0f - w2), c01 = w1 * (1.0f - w2);
  const float c10 = (1.0f - w1) * w2,          c11 = w1 * w2;

  float* __restrict__ o = out + 32ull * (unsigned)p;

#pragma unroll
  for (int cc = 0; cc < CH; cc += 4) {
    // 12 aligned b128 gathers per 4-channel chunk (3 planes x 4 corners).
    v4f f0 = ld4(b0 + cc) * a00 + ld4(b0 + cc + 32) * a01 +
             ld4(b0 + cc + RSTRIDE) * a10 + ld4(b0 + cc + RSTRIDE + 32) * a11;
    v4f f1 = ld4(b1 + cc) * b00 + ld4(b1 + cc + 32) * b01 +
             ld4(b1 + cc + RSTRIDE) * b10 + ld4(b1 + cc + RSTRIDE + 32) * b11;
    v4f f2 = ld4(b2 + cc) * c00 + ld4(b2 + cc + 32) * c01 +
             ld4(b2 + cc + RSTRIDE) * c10 + ld4(b2 + cc + RSTRIDE + 32) * c11;
    v4f r = f0 * f1 * f2;
    __builtin_nontemporal_store(r, (v4f*)(o + cc));  // stream 256MB past L2
  }
}

// ---------- fallback: direct (C,H,W) gather (round-1 kernel) ----------
__global__ void __launch_bounds__(256)
triplane_sample_kernel(const float* __restrict__ x,
                       const float* __restrict__ p0,
                       const float* __restrict__ p1,
                       const float* __restrict__ p2,
                       float* __restrict__ out, int n) {
  int p = blockIdx.x * 256 + threadIdx.x;
  if (p >= n) return;

  const float* __restrict__ xp = x + 3ull * (unsigned)p;
  float g0 = __builtin_nontemporal_load(xp + 0);
  float g1 = __builtin_nontemporal_load(xp + 1);
  float g2 = __builtin_nontemporal_load(xp + 2);

  int c0, c1, c2;
  float w0, w1, w2;
  decomp(g0, c0, w0);
  decomp(g1, c1, w1);
  decomp(g2, c2, w2);

  const float* __restrict__ b0 = p0 + (c1 * RESO + c0);
  const float* __restrict__ b1 = p1 + (c2 * RESO + c0);
  const float* __restrict__ b2 = p2 + (c2 * RESO + c1);

  v2f wA0 = {(1.0f - w0) * (1.0f - w1), w0 * (1.0f - w1)};
  v2f wB0 = {(1.0f - w0) * w1,          w0 * w1};
  v2f wA1 = {(1.0f - w0) * (1.0f - w2), w0 * (1.0f - w2)};
  v2f wB1 = {(1.0f - w0) * w2,          w0 * w2};
  v2f wA2 = {(1.0f - w1) * (1.0f - w2), w1 * (1.0f - w2)};
  v2f wB2 = {(1.0f - w1) * w2,          w1 * w2};

  float* __restrict__ o = out + 32ull * (unsigned)p;

#pragma unroll
  for (int cc = 0; cc < CH; cc += 8) {
    float acc[8];
#pragma unroll
    for (int i = 0; i < 8; ++i) {
      const size_t coff = (size_t)(cc + i) * PLANE_ELEMS;
      v2f t0 = ld2(b0 + coff) * wA0 + ld2(b0 + coff + RESO) * wB0;
      v2f t1 = ld2(b1 + coff) * wA1 + ld2(b1 + coff + RESO) * wB1;
      v2f t2 = ld2(b2 + coff) * wA2 + ld2(b2 + coff + RESO) * wB2;
      acc[i] = (t0.x + t0.y) * (t1.x + t1.y) * (t2.x + t2.y);
    }
    v4f lo = {acc[0], acc[1], acc[2], acc[3]};
    v4f hi = {acc[4], acc[5], acc[6], acc[7]};
    __builtin_nontemporal_store(lo, (v4f*)(o + cc));
    __builtin_nontemporal_store(hi, (v4f*)(o + cc + 4));
  }
}

extern "C" void kernel_launch(void* const* d_in, const int* in_sizes, int n_in,
                              void* d_out, int out_size, void* d_ws, size_t ws_size,
                              hipStream_t stream) {
  const float* x  = (const float*)d_in[0];
  const float* p0 = (const float*)d_in[1];
  const float* p1 = (const float*)d_in[2];
  const float* p2 = (const float*)d_in[3];
  float* out = (float*)d_out;

  int n = in_sizes[0] / 3;             // 2,000,000 points
  int blocks = (n + 255) / 256;

  if (ws_size >= TBYTES) {
    // 3*512*16 tiles / 8 waves per block = 3072 blocks, exact cover.
    transpose_planes_kernel<<<3072, 256, 0, stream>>>(p0, p1, p2, (float*)d_ws);
    triplane_sample_tr_kernel<<<blocks, 256, 0, stream>>>(x, (const float*)d_ws,
                                                          out, n);
  } else {
    triplane_sample_kernel<<<blocks, 256, 0, stream>>>(x, p0, p1, p2, out, n);
  }
}